// CD_1013612282415
// MI455X (gfx1250) — compile-verified
//
#include <hip/hip_runtime.h>

// CDNA5 (gfx1250) Chamfer distance via V_WMMA_F32_16X16X4_F32.
//
// dist(i,j) = |p_i|^2 + |g_j|^2 - 2 p.g
// Augment: A_i = (-2px,-2py,-2pz, 1), B_j = (gx,gy,gz,|g_j|^2)
//   => (A x B)_ij = -2 p.g + |g_j|^2 ; add |p_i|^2 after min over j.
//
// Layouts assumed per CDNA5 ISA §7.12.2:
//  A (16x4 f32, 2 VGPRs): lanes 0-15 -> M=lane, K={0,1}; lanes 16-31 -> M=lane-16, K={2,3}
//  B (4x16 f32, 2 VGPRs): lanes 0-15 -> N=lane, K={0,1}; lanes 16-31 -> N=lane-16, K={2,3}
//  C/D (16x16 f32, 8 VGPRs): VGPR v, lanes 0-15 -> (M=v, N=lane); lanes 16-31 -> (M=v+8, N=lane-16)

typedef float v2f __attribute__((ext_vector_type(2)));
typedef float v8f __attribute__((ext_vector_type(8)));

constexpr int BATCH = 8;
constexpr int NPTS  = 8192;   // N == M
constexpr int CHUNK = 2048;   // gt points staged in LDS per pass chunk (32 KB)
constexpr int WAVES = 8;      // waves per block (wave32)
constexpr int BLOCK = WAVES * 32;
constexpr int ROWS_PER_BLOCK = WAVES * 16;

__global__ __launch_bounds__(BLOCK)
void chamfer_dir_kernel(const float* __restrict__ A,   // "query" points [B][NPTS][3]
                        const float* __restrict__ G,   // "target" points [B][NPTS][3]
                        float* __restrict__ outMin)    // [B][NPTS] min sq-dist per query
{
    __shared__ float4 tile[CHUNK];   // (gx, gy, gz, |g|^2)

    const int b    = blockIdx.y;
    const int lane = threadIdx.x & 31;
    const int wave = threadIdx.x >> 5;
    const int i0   = blockIdx.x * ROWS_PER_BLOCK + wave * 16;

    const float* __restrict__ Ab = A + (size_t)b * NPTS * 3;
    const float* __restrict__ Gb = G + (size_t)b * NPTS * 3;

    // ---- A fragment (augmented, per assumed 16x4 layout) ----
    const int  rowA = i0 + (lane & 15);
    const bool hiK  = (lane >= 16);          // K = {2,3} half
    const float p0 = Ab[3 * rowA + 0];
    const float p1 = Ab[3 * rowA + 1];
    const float p2 = Ab[3 * rowA + 2];
    v2f afrag;
    afrag.x = hiK ? (-2.0f * p2) : (-2.0f * p0);
    afrag.y = hiK ? 1.0f         : (-2.0f * p1);

    float m[8];
    #pragma unroll
    for (int v = 0; v < 8; ++v) m[v] = 3.402823466e+38f;

    const int koff = hiK ? 2 : 0;            // float2 at .xy or .zw of staged float4
    const int col  = lane & 15;

    for (int c0 = 0; c0 < NPTS; c0 += CHUNK) {
        __syncthreads();
        // Cooperative stage: CHUNK target points -> LDS, augmented with |g|^2
        for (int t = threadIdx.x; t < CHUNK; t += BLOCK) {
            const float* g = Gb + 3 * (size_t)(c0 + t);
            const float g0 = g[0], g1 = g[1], g2 = g[2];
            tile[t] = make_float4(g0, g1, g2, g0 * g0 + g1 * g1 + g2 * g2);
        }
        __syncthreads();

        #pragma unroll 4
        for (int jt = 0; jt < CHUNK / 16; ++jt) {
            // ---- B fragment straight out of LDS (contiguous float2) ----
            const float* bp = reinterpret_cast<const float*>(&tile[jt * 16 + col]);
            v2f bfrag;
            bfrag.x = bp[koff + 0];
            bfrag.y = bp[koff + 1];

            v8f cz = {};  // C = 0
            // D = A x B : -2 p.g + |g|^2 for this 16x16 tile
            v8f d = __builtin_amdgcn_wmma_f32_16x16x4_f32(
                false, afrag, false, bfrag, (short)0, cz, false, false);

            #pragma unroll
            for (int v = 0; v < 8; ++v) m[v] = fminf(m[v], d[v]);
        }
    }

    // ---- reduce min over the 16 lanes of each half-wave (columns of the tile) ----
    #pragma unroll
    for (int mask = 1; mask <= 8; mask <<= 1) {
        #pragma unroll
        for (int v = 0; v < 8; ++v)
            m[v] = fminf(m[v], __shfl_xor(m[v], mask, 32));
    }

    // lanes 0 and 16 hold row minima for rows i0+v and i0+8+v; add |p|^2 and store
    if ((lane & 15) == 0) {
        const int rbase = i0 + (hiK ? 8 : 0);
        #pragma unroll
        for (int v = 0; v < 8; ++v) {
            const int r = rbase + v;
            const float q0 = Ab[3 * r + 0];
            const float q1 = Ab[3 * r + 1];
            const float q2 = Ab[3 * r + 2];
            outMin[(size_t)b * NPTS + r] = m[v] + (q0 * q0 + q1 * q1 + q2 * q2);
        }
    }
}

__global__ __launch_bounds__(256)
void chamfer_reduce_kernel(const float* __restrict__ mins, float* __restrict__ out,
                           int n, float scale)
{
    __shared__ float s[256];
    float acc = 0.0f;
    for (int i = threadIdx.x; i < n; i += 256) acc += mins[i];
    s[threadIdx.x] = acc;
    __syncthreads();
    #pragma unroll
    for (int off = 128; off > 0; off >>= 1) {
        if (threadIdx.x < off) s[threadIdx.x] += s[threadIdx.x + off];
        __syncthreads();
    }
    if (threadIdx.x == 0) out[0] = s[0] * scale;
}

extern "C" void kernel_launch(void* const* d_in, const int* in_sizes, int n_in,
                              void* d_out, int out_size, void* d_ws, size_t ws_size,
                              hipStream_t stream)
{
    (void)in_sizes; (void)n_in; (void)out_size; (void)ws_size;
    const float* pred = (const float*)d_in[0];   // [8,8192,3] f32
    const float* gt   = (const float*)d_in[1];   // [8,8192,3] f32
    float* minP = (float*)d_ws;                  // [8*8192]
    float* minG = minP + (size_t)BATCH * NPTS;   // [8*8192] (contiguous after minP)
    float* out  = (float*)d_out;

    dim3 grid(NPTS / ROWS_PER_BLOCK, BATCH);     // (64, 8)
    chamfer_dir_kernel<<<grid, BLOCK, 0, stream>>>(pred, gt, minP); // pred -> nearest gt
    chamfer_dir_kernel<<<grid, BLOCK, 0, stream>>>(gt, pred, minG); // gt -> nearest pred

    // result = (sum(minP)+sum(minG)) * SCALE / (2*B*N)
    const float scale = 100.0f / (2.0f * BATCH * NPTS);
    chamfer_reduce_kernel<<<1, 256, 0, stream>>>(minP, out, 2 * BATCH * NPTS, scale);
}